// SphereTracingRenderer_63788854280577
// MI455X (gfx1250) — compile-verified
//
#include <hip/hip_runtime.h>
#include <hip/hip_bf16.h>
#include <math.h>

// CDNA5 / gfx1250 sphere tracer. Rank-3 MLP contraction done with
// V_WMMA_F32_16X16X4_F32 (exact f32 math, minimal K padding 3->4).
// Round 2: bias moved out of the WMMA C operand (inline-0 SRC2) and the
// hidden-tile loop unrolled only 4x to keep VGPR usage under 256.

typedef __attribute__((ext_vector_type(2))) float v2f;
typedef __attribute__((ext_vector_type(8))) float v8f;

#define HIDDEN    256
#define MAX_ITERS 32
#define FAR_T     6.0f
#define HIT_EPS   1e-5f
#define PERTURB   1e-3f

__device__ __forceinline__ float lane_bcast(float v, int srcLane) {
  // wave32 gather: dst[lane] = v[srcLane]
  return __int_as_float(__builtin_amdgcn_ds_bpermute(srcLane << 2, __float_as_int(v)));
}

__global__ __launch_bounds__(256)
void sphere_trace_wmma_f32(const float* __restrict__ origins,
                           const float* __restrict__ dirs,
                           const float* __restrict__ center,
                           const float* __restrict__ radius,
                           const float* __restrict__ W1,
                           const float* __restrict__ b1,
                           const float* __restrict__ W2,
                           const float* __restrict__ b2,
                           const float* __restrict__ Wc1,
                           const float* __restrict__ bc1,
                           const float* __restrict__ Wc2,
                           const float* __restrict__ bc2,
                           float* __restrict__ out,
                           int nRays) {
  // --- LDS staging (shared by all 8 waves of the block) ---
  // sB[t*32+lane]: B-operand (4x16 f32 tile of W1, K=3 padded to 4) for
  // V_WMMA_F32_16X16X4_F32. Lanes 0-15 hold {W1[0][c], W1[1][c]},
  // lanes 16-31 hold {W1[2][c], 0}, c = t*16 + (lane&15).
  __shared__ float2 sB[16 * 32];
  __shared__ float2 sBW[HIDDEN];      // (b1[j], W2[j])
  __shared__ float4 sC1[HIDDEN];      // (Wc1[0][j], Wc1[1][j], Wc1[2][j], bc1[j])
  __shared__ float4 sC2[HIDDEN];      // (Wc2[j][0], Wc2[j][1], Wc2[j][2], 0)
  __shared__ float  sPert[8][32];     // per-wave pert routing scratch

  const int tid  = threadIdx.x;
  const int lane = tid & 31;
  const int wv   = tid >> 5;

  // Fill W1 B-tiles: 512 entries, 256 threads -> 2 each.
  for (int i = tid; i < 16 * 32; i += 256) {
    const int ln = i & 31;
    const int c  = ((i >> 5) << 4) + (ln & 15);
    float2 v;
    if (ln < 16) v = make_float2(W1[0 * HIDDEN + c], W1[1 * HIDDEN + c]);
    else         v = make_float2(W1[2 * HIDDEN + c], 0.0f);
    sB[i] = v;
  }
  {
    const int j = tid;   // blockDim == HIDDEN == 256
    sBW[j] = make_float2(b1[j], W2[j]);
    sC1[j] = make_float4(Wc1[0 * HIDDEN + j], Wc1[1 * HIDDEN + j],
                         Wc1[2 * HIDDEN + j], bc1[j]);
    sC2[j] = make_float4(Wc2[3 * j + 0], Wc2[3 * j + 1], Wc2[3 * j + 2], 0.0f);
  }
  __syncthreads();

  // --- per-lane ray state ---
  const int rid = blockIdx.x * 256 + tid;   // nRays is a multiple of 256
  float px = origins[3 * rid + 0];
  float py = origins[3 * rid + 1];
  float pz = origins[3 * rid + 2];
  const float dx = dirs[3 * rid + 0];
  const float dy = dirs[3 * rid + 1];
  const float dz = dirs[3 * rid + 2];

  const float cx = center[0], cy = center[1], cz = center[2];
  const float rad = radius[0];
  const float b2v = b2[0];

  float dist = 0.0f;
  bool unfinished = true;
  bool hit = false;

  const int  sl    = lane & 15;
  const bool lower = lane < 16;

#pragma unroll 1
  for (int it = 0; it < MAX_ITERS; ++it) {
    // Build two A operands (16x4 f32): rays 0-15 and rays 16-31 of this wave.
    // f32 A layout: lanes 0-15 VGPR{0,1} = (K0,K1) of row M=lane;
    //               lanes 16-31 VGPR{0,1} = (K2,K3) of row M=lane-16.
    const float xs  = lane_bcast(px, sl);
    const float ys  = lane_bcast(py, sl);
    const float zs  = lane_bcast(pz, sl);
    const float xs2 = lane_bcast(px, sl + 16);
    const float ys2 = lane_bcast(py, sl + 16);
    const float zs2 = lane_bcast(pz, sl + 16);

    v2f a0, a1;
    a0[0] = lower ? xs  : zs;   a0[1] = lower ? ys  : 0.0f;
    a1[0] = lower ? xs2 : zs2;  a1[1] = lower ? ys2 : 0.0f;

    float acc0[8], acc1[8];
#pragma unroll
    for (int r = 0; r < 8; ++r) { acc0[r] = 0.0f; acc1[r] = 0.0f; }

    // H = P @ W1tile (bias folded into epilogue; C operand is inline 0 so
    // no VGPRs are burned on broadcast bias tiles). Then fused
    // relu(H + b1) * W2 accumulation. Unroll 4 to bound live D tiles.
#pragma unroll 4
    for (int t = 0; t < 16; ++t) {
      const float2 bw = sBW[t * 16 + sl];     // (b1, w2) for this lane's column
      const float2 bl = sB[t * 32 + lane];
      v2f bmat; bmat[0] = bl.x; bmat[1] = bl.y;
      const v8f czero = {};                   // lowers to inline-0 SRC2

      v8f d0 = __builtin_amdgcn_wmma_f32_16x16x4_f32(
          false, a0, false, bmat, (short)0, czero, false, false);
      v8f d1 = __builtin_amdgcn_wmma_f32_16x16x4_f32(
          false, a1, false, bmat, (short)0, czero, false, false);

#pragma unroll
      for (int r = 0; r < 8; ++r) {
        acc0[r] = fmaf(fmaxf(d0[r] + bw.x, 0.0f), bw.y, acc0[r]);
        acc1[r] = fmaf(fmaxf(d1[r] + bw.x, 0.0f), bw.y, acc1[r]);
      }
    }

    // Butterfly-reduce across the 16 lanes of each half (N dimension).
#pragma unroll
    for (int r = 0; r < 8; ++r) {
      float v0 = acc0[r], v1 = acc1[r];
#pragma unroll
      for (int m = 1; m <= 8; m <<= 1) {
        v0 += lane_bcast(v0, lane ^ m);
        v1 += lane_bcast(v1, lane ^ m);
      }
      acc0[r] = v0; acc1[r] = v1;
    }

    // Route pert[ray] back to owning lane via per-wave LDS scratch.
    // acc0: lower half -> rays 0-7, upper half -> rays 8-15.
    // acc1: lower half -> rays 16-23, upper half -> rays 24-31.
    if (lane == 0 || lane == 16) {
      const int base = (lane >> 4) * 8;
#pragma unroll
      for (int r = 0; r < 8; ++r) {
        sPert[wv][base + r]      = acc0[r];
        sPert[wv][16 + base + r] = acc1[r];
      }
    }
    // Same-wave LDS ops are in-order; compiler tracks the memory dependency.
    const float pert = sPert[wv][lane];

    // SDF = analytic sphere + bounded neural perturbation.
    const float ox = px - cx, oy = py - cy, oz = pz - cz;
    const float base = sqrtf(ox * ox + oy * oy + oz * oz) - rad;
    const float dval = base + PERTURB * tanhf(pert + b2v);

    const bool curr_hit = fabsf(dval) < HIT_EPS;
    const bool missed   = dist > FAR_T;          // checked BEFORE accumulation
    const bool still    = unfinished && !curr_hit && !missed;
    hit        = hit || (unfinished && curr_hit);
    unfinished = unfinished && !curr_hit && !missed;

    const float upd = still ? dval : 0.0f;
    px += upd * dx; py += upd * dy; pz += upd * dz;
    dist += upd;

    if (__builtin_amdgcn_ballot_w32(unfinished) == 0u) break;  // wave-uniform
  }

  // --- color MLP (one shot, per hit lane) ---
  float colr = 0.0f, colg = 0.0f, colb = 0.0f;
  if (hit) {
    float cr = 0.0f, cg = 0.0f, cb = 0.0f;
#pragma unroll 4
    for (int j = 0; j < HIDDEN; ++j) {
      const float4 w = sC1[j];   // wave-uniform LDS broadcast
      const float h = fmaxf(fmaf(px, w.x, fmaf(py, w.y, fmaf(pz, w.z, w.w))), 0.0f);
      const float4 u = sC2[j];
      cr = fmaf(h, u.x, cr);
      cg = fmaf(h, u.y, cg);
      cb = fmaf(h, u.z, cb);
    }
    colr = 1.0f / (1.0f + expf(-(cr + bc2[0])));
    colg = 1.0f / (1.0f + expf(-(cg + bc2[1])));
    colb = 1.0f / (1.0f + expf(-(cb + bc2[2])));
  }
  out[3 * rid + 0] = colr;
  out[3 * rid + 1] = colg;
  out[3 * rid + 2] = colb;
}

extern "C" void kernel_launch(void* const* d_in, const int* in_sizes, int n_in,
                              void* d_out, int out_size, void* d_ws, size_t ws_size,
                              hipStream_t stream) {
  (void)n_in; (void)d_ws; (void)ws_size; (void)out_size;
  const float* origins = (const float*)d_in[0];
  const float* dirs    = (const float*)d_in[1];
  const float* center  = (const float*)d_in[2];
  const float* radius  = (const float*)d_in[3];
  const float* W1      = (const float*)d_in[4];
  const float* b1      = (const float*)d_in[5];
  const float* W2      = (const float*)d_in[6];
  const float* b2      = (const float*)d_in[7];
  const float* Wc1     = (const float*)d_in[8];
  const float* bc1     = (const float*)d_in[9];
  const float* Wc2     = (const float*)d_in[10];
  const float* bc2     = (const float*)d_in[11];
  float* out = (float*)d_out;

  const int nRays  = in_sizes[0] / 3;
  const int blocks = (nRays + 255) / 256;
  hipLaunchKernelGGL(sphere_trace_wmma_f32, dim3(blocks), dim3(256), 0, stream,
                     origins, dirs, center, radius, W1, b1, W2, b2,
                     Wc1, bc1, Wc2, bc2, out, nRays);
}